// DepthProjectionModule_69312182223026
// MI455X (gfx1250) — compile-verified
//
#include <hip/hip_runtime.h>
#include <hip/hip_bf16.h>

#define W_IMG 320
#define H_IMG 256
#define B_N   8
#define D_N   64
#define WH    (W_IMG * H_IMG)

typedef __attribute__((ext_vector_type(2))) float v2f;
typedef __attribute__((ext_vector_type(8))) float v8f;

// Fast reciprocal: v_rcp_f32 (TRANS, co-executes with VALU) + one NR step.
__device__ __forceinline__ float fast_rcp(float x) {
    float r = __builtin_amdgcn_rcpf(x);
    r = fmaf(fmaf(-x, r, 1.0f), r, r);   // ~1 ulp
    return r;
}

// Wave32 tile: 16 consecutive w-pixels (one row) x 64 depths.
// Coordinate generation T[d,p] = depth[d]*G[comp,p] + KT[comp] is done with
// V_WMMA_F32_16X16X4_F32 (A = [depth,1,0,0] per depth-row, B = [G;KT;0;0]).
// A-matrix layout (ISA 7.12.2, 32-bit 16x4): lanes0-15 hold K=0,1; lanes16-31
// hold K=2,3 -> zeroing upper lanes makes B's K=2,3 rows irrelevant.
__launch_bounds__(256)
__global__ void depthproj_costvol_wmma(const float* __restrict__ left,
                                       const float* __restrict__ right,
                                       const float* __restrict__ krkiuv,
                                       const float* __restrict__ kt,
                                       float* __restrict__ out)
{
    const float IDB = 1.0f / 50.0f;
    const float IDS = (2.0f - 1.0f / 50.0f) / 63.0f;

    const int lane = threadIdx.x & 31;
    const int wid  = blockIdx.x * (blockDim.x >> 5) + (threadIdx.x >> 5);
    const int TPR  = W_IMG / 16;        // 20 tiles per image row
    const int WPB  = H_IMG * TPR;       // waves per batch
    const int b    = wid / WPB;
    const int rr   = wid - b * WPB;
    const int h    = rr / TPR;
    const int w0   = (rr - h * TPR) * 16;

    const int  p  = lane & 15;          // pixel within tile (N index)
    const bool hi = lane >= 16;
    const int  w  = w0 + p;
    const int  g  = w * H_IMG + h;      // geometry column: flat index is w*H + h

    // B operands (one per homography component), constant across depth blocks.
    const float* Gb = krkiuv + (size_t)b * 3 * WH;
    const float* Tb = kt     + (size_t)b * 3;
    v2f Bm0, Bm1, Bm2;
    {
        float g0 = Gb[0 * WH + g], g1 = Gb[1 * WH + g], g2 = Gb[2 * WH + g];
        float t0 = Tb[0],          t1 = Tb[1],          t2 = Tb[2];
        Bm0.x = hi ? 0.0f : g0;  Bm0.y = hi ? 0.0f : t0;
        Bm1.x = hi ? 0.0f : g1;  Bm1.y = hi ? 0.0f : t1;
        Bm2.x = hi ? 0.0f : g2;  Bm2.y = hi ? 0.0f : t2;
    }

    // Left-image pixel (3 channels) lives in registers for all 64 depths.
    const float* Lb = left + (size_t)(b * 3) * WH + (size_t)h * W_IMG + w;
    const float l0 = Lb[0], l1 = Lb[WH], l2 = Lb[2 * WH];

    const float* Rb = right + (size_t)(b * 3) * WH;
    float* Ob = out + (size_t)(b * D_N) * WH + (size_t)h * W_IMG + w;

    const float sx = (W_IMG - 1.0f) / (float)W_IMG;  // ix = u*(W-1)/W
    const float sy = (H_IMG - 1.0f) / (float)H_IMG;  // iy = v*(H-1)/H

    #pragma unroll 1
    for (int kb = 0; kb < 4; ++kb) {
        // A: row M=d holds [depth(kb*16+M), 1] in K=0,1; K=2,3 rows zero.
        float dep = fast_rcp(IDB + (float)(kb * 16 + p) * IDS);
        v2f Am;
        Am.x = hi ? 0.0f : dep;
        Am.y = hi ? 0.0f : 1.0f;
        v8f cz = {0.f, 0.f, 0.f, 0.f, 0.f, 0.f, 0.f, 0.f};

        v8f U = __builtin_amdgcn_wmma_f32_16x16x4_f32(false, Am, false, Bm0,
                                                      (short)0, cz, false, false);
        v8f V = __builtin_amdgcn_wmma_f32_16x16x4_f32(false, Am, false, Bm1,
                                                      (short)0, cz, false, false);
        v8f Z = __builtin_amdgcn_wmma_f32_16x16x4_f32(false, Am, false, Bm2,
                                                      (short)0, cz, false, false);

        #pragma unroll
        for (int r = 0; r < 8; ++r) {
            const int d = kb * 16 + r + (hi ? 8 : 0);  // D layout: M = r + 8*(lane>=16)

            float zc  = Z[r] + 1e-6f;
            float inv = fast_rcp(zc);
            float ix  = U[r] * inv * sx;
            float iy  = V[r] * inv * sy;

            float x0f = floorf(ix), y0f = floorf(iy);
            float wx1 = ix - x0f,   wy1 = iy - y0f;
            float wx0 = 1.0f - wx1, wy0 = 1.0f - wy1;
            float x1f = x0f + 1.0f, y1f = y0f + 1.0f;

            bool vx0 = (x0f >= 0.0f) && (x0f <= (float)(W_IMG - 1));
            bool vx1 = (x1f >= 0.0f) && (x1f <= (float)(W_IMG - 1));
            bool vy0 = (y0f >= 0.0f) && (y0f <= (float)(H_IMG - 1));
            bool vy1 = (y1f >= 0.0f) && (y1f <= (float)(H_IMG - 1));

            int xi0 = (int)fminf(fmaxf(x0f, 0.0f), (float)(W_IMG - 1));
            int xi1 = (int)fminf(fmaxf(x1f, 0.0f), (float)(W_IMG - 1));
            int yi0 = (int)fminf(fmaxf(y0f, 0.0f), (float)(H_IMG - 1));
            int yi1 = (int)fminf(fmaxf(y1f, 0.0f), (float)(H_IMG - 1));

            float w00 = wx0 * wy0 * ((vx0 && vy0) ? 1.0f : 0.0f);
            float w10 = wx1 * wy0 * ((vx1 && vy0) ? 1.0f : 0.0f);
            float w01 = wx0 * wy1 * ((vx0 && vy1) ? 1.0f : 0.0f);
            float w11 = wx1 * wy1 * ((vx1 && vy1) ? 1.0f : 0.0f);

            int i00 = yi0 * W_IMG + xi0, i10 = yi0 * W_IMG + xi1;
            int i01 = yi1 * W_IMG + xi0, i11 = yi1 * W_IMG + xi1;

            float s0 = w00 * Rb[i00]          + w10 * Rb[i10]
                     + w01 * Rb[i01]          + w11 * Rb[i11];
            float s1 = w00 * Rb[WH + i00]     + w10 * Rb[WH + i10]
                     + w01 * Rb[WH + i01]     + w11 * Rb[WH + i11];
            float s2 = w00 * Rb[2 * WH + i00] + w10 * Rb[2 * WH + i10]
                     + w01 * Rb[2 * WH + i01] + w11 * Rb[2 * WH + i11];

            float cost = fabsf(s0 - l0) + fabsf(s1 - l1) + fabsf(s2 - l2);

            // Output is write-once (168 MB): non-temporal so it doesn't evict
            // the L2-resident images that the gathers keep re-reading.
            __builtin_nontemporal_store(cost, Ob + (size_t)d * WH);
        }
    }
}

extern "C" void kernel_launch(void* const* d_in, const int* in_sizes, int n_in,
                              void* d_out, int out_size, void* d_ws, size_t ws_size,
                              hipStream_t stream) {
    (void)in_sizes; (void)n_in; (void)out_size; (void)d_ws; (void)ws_size;
    const float* left   = (const float*)d_in[0];
    const float* right  = (const float*)d_in[1];
    const float* krkiuv = (const float*)d_in[2];
    const float* kt     = (const float*)d_in[3];
    float* out = (float*)d_out;

    // waves = B * H * (W/16) = 8*256*20 = 40960 ; 8 waves per 256-thread block
    dim3 block(256);
    dim3 grid((B_N * H_IMG * (W_IMG / 16)) / 8);   // 5120 blocks, exact cover
    hipLaunchKernelGGL(depthproj_costvol_wmma, grid, block, 0, stream,
                       left, right, krkiuv, kt, out);
}